// GraphAttentionLayer_67740224193140
// MI455X (gfx1250) — compile-verified
//
#include <hip/hip_runtime.h>
#include <hip/hip_bf16.h>

#define ALPHA 0.2f
#define NEG_INF -9000000000000000.0f

typedef __attribute__((ext_vector_type(2))) float v2f;
typedef __attribute__((ext_vector_type(8))) float v8f;

// ---------------------------------------------------------------------------
// C[M x N] = act( A[M x K] @ B[K x N] ), all row-major fp32.
// One wave computes one 16x16 tile of C using V_WMMA_F32_16X16X4_F32.
// act: 0 = identity, 1 = elu
// ---------------------------------------------------------------------------
__global__ __launch_bounds__(128) void wmma_gemm_f32_kernel(
    const float* __restrict__ A, const float* __restrict__ B,
    float* __restrict__ C, int M, int N, int K, int act) {
  const int waveInBlock = threadIdx.x >> 5;
  const int lane = threadIdx.x & 31;
  const int tilesN = N >> 4;
  const int totalTiles = (M >> 4) * tilesN;
  const int tile = blockIdx.x * 4 + waveInBlock;
  if (tile >= totalTiles) return;  // wave-uniform: EXEC stays all-1s

  const int tm = (tile / tilesN) << 4;
  const int tn = (tile % tilesN) << 4;
  const int half = lane >> 4;  // 0: K pair {0,1}; 1: K pair {2,3}
  const int l16 = lane & 15;

  // A: lane holds (M = tm+l16, K = k + half*2 .. +1) -> aligned float2 load
  const float* arow = A + (size_t)(tm + l16) * K + (half << 1);
  // B: VGPR0 = row K=k+half*2, VGPR1 = row K=k+half*2+1, N = tn+l16
  const float* bcol = B + tn + l16;

  v8f c = {};
#pragma unroll 4
  for (int k = 0; k < K; k += 4) {
    v2f a = *(const v2f*)(arow + k);
    v2f b;
    b.x = bcol[(size_t)(k + (half << 1)) * N];
    b.y = bcol[(size_t)(k + (half << 1) + 1) * N];
    c = __builtin_amdgcn_wmma_f32_16x16x4_f32(
        /*neg_a=*/false, a, /*neg_b=*/false, b,
        /*c_mod=*/(short)0, c, /*reuse_a=*/false, /*reuse_b=*/false);
  }

  // C/D layout: VGPR r -> M = tm + r + half*8, N = tn + l16
  float* crow = C + (size_t)(tm + (half << 3)) * N + tn + l16;
#pragma unroll
  for (int r = 0; r < 8; ++r) {
    float v = c[r];
    if (act) v = v > 0.f ? v : (__expf(v) - 1.f);
    crow[(size_t)r * N] = v;
  }
}

// ---------------------------------------------------------------------------
// s1[i] = dot(Wh[i,:], a[0:128]); s2[i] = dot(Wh[i,:], a[128:256])
// One wave per row, float4 lane loads + xor-shuffle reduction.
// ---------------------------------------------------------------------------
__global__ __launch_bounds__(256) void s12_kernel(
    const float* __restrict__ Wh, const float* __restrict__ a,
    float* __restrict__ s1, float* __restrict__ s2) {
  const int wave = (blockIdx.x * blockDim.x + threadIdx.x) >> 5;
  const int lane = threadIdx.x & 31;
  if (wave >= 1024) return;
  float4 w = ((const float4*)(Wh + (size_t)wave * 128))[lane];
  float4 a1 = ((const float4*)(a))[lane];
  float4 a2 = ((const float4*)(a + 128))[lane];
  float p1 = w.x * a1.x + w.y * a1.y + w.z * a1.z + w.w * a1.w;
  float p2 = w.x * a2.x + w.y * a2.y + w.z * a2.z + w.w * a2.w;
#pragma unroll
  for (int m = 16; m >= 1; m >>= 1) {
    p1 += __shfl_xor(p1, m, 32);
    p2 += __shfl_xor(p2, m, 32);
  }
  if (lane == 0) { s1[wave] = p1; s2[wave] = p2; }
}

// ---------------------------------------------------------------------------
// wa3[i] = sum_k W[i,k] * a[256+k]   (tiny 128x128 matvec)
// ---------------------------------------------------------------------------
__global__ __launch_bounds__(128) void wa3_kernel(
    const float* __restrict__ W, const float* __restrict__ a,
    float* __restrict__ wa3) {
  const int i = threadIdx.x;
  float acc = 0.f;
  for (int k = 0; k < 128; ++k) acc += W[(size_t)i * 128 + k] * a[256 + k];
  wa3[i] = acc;
}

// ---------------------------------------------------------------------------
// Fused per-row GAT: for row i
//   e[j]     = leakyrelu(s1[i] + s2[j] + dot(n2n[i*N+j,:], wa3))
//   masked   = adjacency ? e : NEG_INF
//   att      = softmax_j(masked)
//   hprime[i,:] = elu(att @ Wh)
// One workgroup (256 thr = 8 waves) per row. Streams 512KB of n2n per row.
// ---------------------------------------------------------------------------
__global__ __launch_bounds__(256) void gat_row_kernel(
    const float* __restrict__ n2n, const int* __restrict__ adj,
    const float* __restrict__ Wh, const float* __restrict__ s1,
    const float* __restrict__ s2, const float* __restrict__ wa3,
    float* __restrict__ hprime) {
  __shared__ float e_sh[1024];
  __shared__ float red[256];

  const int row = blockIdx.x;
  const int t = threadIdx.x;
  const int wave = t >> 5;
  const int lane = t & 31;

  const float4 w3 = ((const float4*)wa3)[lane];
  const float s1r = s1[row];
  const float* base = n2n + (size_t)row * 1024 * 128;

  // ---- Phase A: e values (2 rows per wave per iter, contiguous 8KB/block/iter)
  float wmax = NEG_INF;
  for (int j0 = wave * 2; j0 < 1024; j0 += 16) {
    float4 x0 = ((const float4*)(base + (size_t)j0 * 128))[lane];
    float4 x1 = ((const float4*)(base + (size_t)(j0 + 1) * 128))[lane];
    float p0 = x0.x * w3.x + x0.y * w3.y + x0.z * w3.z + x0.w * w3.w;
    float p1 = x1.x * w3.x + x1.y * w3.y + x1.z * w3.z + x1.w * w3.w;
#pragma unroll
    for (int m = 16; m >= 1; m >>= 1) {
      p0 += __shfl_xor(p0, m, 32);
      p1 += __shfl_xor(p1, m, 32);
    }
    float e0 = s1r + s2[j0] + p0;
    float e1 = s1r + s2[j0 + 1] + p1;
    e0 = e0 > 0.f ? e0 : ALPHA * e0;
    e1 = e1 > 0.f ? e1 : ALPHA * e1;
    e0 = (adj[row * 1024 + j0] > 0) ? e0 : NEG_INF;
    e1 = (adj[row * 1024 + j0 + 1] > 0) ? e1 : NEG_INF;
    if (lane == 0) { e_sh[j0] = e0; e_sh[j0 + 1] = e1; }
    wmax = fmaxf(wmax, fmaxf(e0, e1));
  }

  // ---- Phase B: row max
  red[t] = wmax;
  __syncthreads();
#pragma unroll
  for (int s = 128; s >= 1; s >>= 1) {
    if (t < s) red[t] = fmaxf(red[t], red[t + s]);
    __syncthreads();
  }
  const float rowmax = red[0];
  __syncthreads();

  // ---- Phase C: exp + row sum
  float lsum = 0.f;
  for (int j = t; j < 1024; j += 256) {
    float p = __expf(e_sh[j] - rowmax);
    e_sh[j] = p;
    lsum += p;
  }
  red[t] = lsum;
  __syncthreads();
#pragma unroll
  for (int s = 128; s >= 1; s >>= 1) {
    if (t < s) red[t] += red[t + s];
    __syncthreads();
  }
  const float inv = 1.0f / red[0];
  __syncthreads();

  // ---- Phase D: hprime[row,:] = elu( (p/sum) @ Wh )
  const int f = t & 127;
  const int g = t >> 7;  // 0 or 1: split j range over two half-blocks
  float acc = 0.f;
  for (int j = g; j < 1024; j += 2) acc += e_sh[j] * Wh[(size_t)j * 128 + f];
  red[t] = acc;
  __syncthreads();
  if (t < 128) {
    float v = (red[t] + red[t + 128]) * inv;
    v = v > 0.f ? v : (__expf(v) - 1.f);
    hprime[(size_t)row * 128 + f] = v;
  }
}

extern "C" void kernel_launch(void* const* d_in, const int* in_sizes, int n_in,
                              void* d_out, int out_size, void* d_ws, size_t ws_size,
                              hipStream_t stream) {
  const float* h         = (const float*)d_in[0];  // 1024 x 128
  const float* edge_attr = (const float*)d_in[1];  // 32768 x 128
  const int*   adjacency = (const int*)d_in[2];    // 1024 x 1024
  const float* n2n       = (const float*)d_in[3];  // (1024*1024) x 128
  const float* W         = (const float*)d_in[4];  // 128 x 128
  const float* a         = (const float*)d_in[5];  // 384 x 1

  const int N = 1024, E = 32768, F = 128;

  float* out      = (float*)d_out;
  float* hprime   = out;             // N * F
  float* edge_out = out + N * F;     // E * F

  float* Wh  = (float*)d_ws;         // N * F floats
  float* s1v = Wh + (size_t)N * F;   // N
  float* s2v = s1v + N;              // N
  float* wa3 = s2v + N;              // F

  // Wh = h @ W  (512 16x16 tiles, 4 waves/block -> 128 blocks)
  wmma_gemm_f32_kernel<<<(N / 16) * (F / 16) / 4, 128, 0, stream>>>(
      h, W, Wh, N, F, F, /*act=*/0);

  // edge_out = elu(edge_attr @ W)  (16384 tiles -> 4096 blocks)
  wmma_gemm_f32_kernel<<<(E / 16) * (F / 16) / 4, 128, 0, stream>>>(
      edge_attr, W, edge_out, E, F, F, /*act=*/1);

  // s1, s2 from Wh (1024 waves -> 128 blocks of 8 waves)
  s12_kernel<<<128, 256, 0, stream>>>(Wh, a, s1v, s2v);

  // wa3 = W @ a3
  wa3_kernel<<<1, 128, 0, stream>>>(W, a, wa3);

  // fused s3 / mask / softmax / attention@Wh — one block per row
  gat_row_kernel<<<N, 256, 0, stream>>>(n2n, adjacency, Wh, s1v, s2v, wa3, hprime);
}